// VariationalAutoEncoder_21208548507786
// MI455X (gfx1250) — compile-verified
//
#include <hip/hip_runtime.h>
#include <hip/hip_bf16.h>

#define H 64
#define NNODE 50000
#define NEDGE 800000
#define EE (NEDGE + NNODE)
#define NB 500
#define HD 256
#define NP 2016
#define OUTD 4032

typedef __attribute__((ext_vector_type(2))) float v2f;
typedef __attribute__((ext_vector_type(8))) float v8f;

// ---------------- fp32 WMMA GEMM: out = A(MxK) @ W(KxNcol) + bias ----------------
// Block = 128 threads (4 waves). Block tile = 64 rows x 64 cols.
// W[:, n0:n0+64] staged in LDS in K-PAIRED layout:
//     lds[(k>>1)*128 + 2*c + (k&1)] = W[k][n0+c]
// so a WMMA B fragment (W[ka][c], W[ka+1][c]) is one aligned ds_load_b64.
// Each wave owns a 16-row strip with 4 accumulators (16x64): 4 WMMAs per A load.
// A fragment is software-pipelined (load k0+4 while computing k0).
// A-frag layout (ISA 7.12.2): lanes 0-15 -> K=ka,ka+1 ; lanes 16-31 -> K=ka+2,ka+3.
__global__ void wmma_gemm_bias(const float* __restrict__ A, const float* __restrict__ W,
                               const float* __restrict__ bias, float* __restrict__ out,
                               int M, int K, int Ncol) {
    extern __shared__ float lds[];          // K/2 x 128 paired block of W
    const int tid  = threadIdx.x;
    const int wave = tid >> 5;
    const int lane = tid & 31;
    const int half = lane >> 4;
    const int l16  = lane & 15;
    const int n0   = blockIdx.y * 64;

    // cooperative stage of W block into LDS (paired layout)
    for (int idx = tid; idx < K * 64; idx += 128) {
        int k = idx >> 6, c = idx & 63;
        lds[(k >> 1) * 128 + 2 * c + (k & 1)] = W[(long)k * Ncol + n0 + c];
    }
    __syncthreads();

    const int rowBase = blockIdx.x * 64 + wave * 16;
    const int row     = rowBase + l16;
    const int rowc    = (row < M) ? row : (M - 1);   // clamp: padded rows never stored
    const float* Ar   = A + (long)rowc * K + 2 * half;

    v8f acc0 = {}, acc1 = {}, acc2 = {}, acc3 = {};
    v2f a;
    a.x = Ar[0];
    a.y = Ar[1];
    for (int k0 = 0; k0 < K; k0 += 4) {
        // prefetch next A fragment (overlaps DS loads + WMMAs below)
        v2f an;
        const int kn = (k0 + 4 < K) ? (k0 + 4) : 0;
        an.x = Ar[kn];
        an.y = Ar[kn + 1];

        // B fragments: contiguous K-pairs -> single ds_load_b64 each
        const float* l0 = lds + ((k0 >> 1) + half) * 128 + 2 * l16;
        v2f b0 = *(const v2f*)(l0 + 0);
        v2f b1 = *(const v2f*)(l0 + 32);
        v2f b2 = *(const v2f*)(l0 + 64);
        v2f b3 = *(const v2f*)(l0 + 96);

        acc0 = __builtin_amdgcn_wmma_f32_16x16x4_f32(false, a, false, b0, (short)0, acc0, false, false);
        acc1 = __builtin_amdgcn_wmma_f32_16x16x4_f32(false, a, false, b1, (short)0, acc1, false, false);
        acc2 = __builtin_amdgcn_wmma_f32_16x16x4_f32(false, a, false, b2, (short)0, acc2, false, false);
        acc3 = __builtin_amdgcn_wmma_f32_16x16x4_f32(false, a, false, b3, (short)0, acc3, false, false);
        a = an;
    }

    const int c0 = n0 + l16;
    const float bv0 = bias ? bias[c0]      : 0.0f;
    const float bv1 = bias ? bias[c0 + 16] : 0.0f;
    const float bv2 = bias ? bias[c0 + 32] : 0.0f;
    const float bv3 = bias ? bias[c0 + 48] : 0.0f;
#pragma unroll
    for (int j = 0; j < 8; ++j) {
        const int m = rowBase + j + 8 * half;   // D layout: VGPR j -> M = j + 8*half
        if (m < M) {
            float* o = out + (long)m * Ncol + c0;
            o[0]  = acc0[j] + bv0;
            o[16] = acc1[j] + bv1;
            o[32] = acc2[j] + bv2;
            o[48] = acc3[j] + bv3;
        }
    }
}

// ---------------- helpers ----------------
__device__ __forceinline__ int f_enc(float f) {
    int i = __float_as_int(f);
    return (i >= 0) ? i : (i ^ 0x7fffffff);   // monotone float->int key
}
__device__ __forceinline__ float f_dec(int k) {
    int i = (k >= 0) ? k : (k ^ 0x7fffffff);
    return __int_as_float(i);
}

__global__ void zero_f(float* p, long n) {
    long i = (long)blockIdx.x * blockDim.x + threadIdx.x;
    if (i < n) p[i] = 0.0f;
}
__global__ void init_i(int* p, long n, int v) {
    long i = (long)blockIdx.x * blockDim.x + threadIdx.x;
    if (i < n) p[i] = v;
}

// ---------------- GAT edge passes ----------------
__global__ void edge_score(const float* __restrict__ hl, const float* __restrict__ hr,
                           const float* __restrict__ att, const int* __restrict__ ei,
                           float* __restrict__ score, int* __restrict__ smax) {
    int e = blockIdx.x * blockDim.x + threadIdx.x;
    if (e >= EE) return;
    int s = (e < NEDGE) ? ei[e]         : (e - NEDGE);
    int d = (e < NEDGE) ? ei[NEDGE + e] : (e - NEDGE);
    const float4* hl4 = (const float4*)(hl + (long)s * H);
    const float4* hr4 = (const float4*)(hr + (long)d * H);
    const float4* at4 = (const float4*)att;
    float sc = 0.0f;
#pragma unroll 4
    for (int j = 0; j < H / 4; ++j) {
        float4 a = hl4[j], b = hr4[j], w = at4[j];   // global_load_b128
        float m0 = a.x + b.x; m0 = (m0 >= 0.0f) ? m0 : 0.2f * m0;
        float m1 = a.y + b.y; m1 = (m1 >= 0.0f) ? m1 : 0.2f * m1;
        float m2 = a.z + b.z; m2 = (m2 >= 0.0f) ? m2 : 0.2f * m2;
        float m3 = a.w + b.w; m3 = (m3 >= 0.0f) ? m3 : 0.2f * m3;
        sc += m0 * w.x + m1 * w.y + m2 * w.z + m3 * w.w;
    }
    score[e] = sc;
    atomicMax(&smax[d], f_enc(sc));
}

__global__ void edge_exp(const float* __restrict__ score, const int* __restrict__ smax,
                         const int* __restrict__ ei, float* __restrict__ ex,
                         float* __restrict__ denom) {
    int e = blockIdx.x * blockDim.x + threadIdx.x;
    if (e >= EE) return;
    int d = (e < NEDGE) ? ei[NEDGE + e] : (e - NEDGE);
    float v = expf(score[e] - f_dec(smax[d]));
    ex[e] = v;
    atomicAdd(&denom[d], v);
}

__global__ void edge_scatter(const float* __restrict__ ex, const float* __restrict__ hl,
                             const int* __restrict__ ei, float* __restrict__ acc) {
    int e = blockIdx.x * blockDim.x + threadIdx.x;
    if (e >= EE) return;
    int s = (e < NEDGE) ? ei[e]         : (e - NEDGE);
    int d = (e < NEDGE) ? ei[NEDGE + e] : (e - NEDGE);
    float w = ex[e];
    const float4* hl4 = (const float4*)(hl + (long)s * H);
    float* ad = acc + (long)d * H;
#pragma unroll 4
    for (int j = 0; j < H / 4; ++j) {
        float4 a = hl4[j];                            // global_load_b128
        atomicAdd(&ad[4 * j + 0], w * a.x);
        atomicAdd(&ad[4 * j + 1], w * a.y);
        atomicAdd(&ad[4 * j + 2], w * a.z);
        atomicAdd(&ad[4 * j + 3], w * a.w);
    }
}

__global__ void node_combine(const float* __restrict__ acc, const float* __restrict__ denom,
                             const float* __restrict__ bo, float* __restrict__ h) {
    long idx = (long)blockIdx.x * blockDim.x + threadIdx.x;
    if (idx >= (long)NNODE * H) return;
    int i = (int)(idx >> 6), j = (int)(idx & 63);
    h[idx] = acc[idx] / (denom[i] + 1e-16f) + bo[j];
}

// ---------------- GraphNorm (global over rows) ----------------
// red layout: [0,f) sum  [f,2f) sumsq  [2f,3f) mean  [3f,4f) invstd
__global__ void reduce_sums(const float* __restrict__ v, int rows, int feats,
                            float* __restrict__ red) {
    int j = threadIdx.x;                  // blockDim.x == feats
    float ls = 0.0f, lss = 0.0f;
    for (int i = blockIdx.x; i < rows; i += gridDim.x) {
        float x = v[(long)i * feats + j];
        ls += x; lss += x * x;
    }
    atomicAdd(&red[j], ls);
    atomicAdd(&red[feats + j], lss);
}

__global__ void norm_stats(float* __restrict__ red, int rows, int feats,
                           const float* __restrict__ ms) {
    int j = threadIdx.x;
    if (j >= feats) return;
    float inv_n = 1.0f / (float)rows;
    float mean  = red[j] * inv_n;
    float m     = ms[j];
    // var of (v - ms*mean): E[v^2] - 2*ms*mean*E[v] + ms^2*mean^2
    float var = red[feats + j] * inv_n - 2.0f * m * mean * mean + m * m * mean * mean;
    red[2 * feats + j] = mean;
    red[3 * feats + j] = rsqrtf(var + 1e-5f);
}

__global__ void normalize_k(float* __restrict__ v, int rows, int feats,
                            const float* __restrict__ w, const float* __restrict__ b,
                            const float* __restrict__ ms, const float* __restrict__ red,
                            const float* __restrict__ prelu_a) {
    long idx = (long)blockIdx.x * blockDim.x + threadIdx.x;
    if (idx >= (long)rows * feats) return;
    int j = (int)(idx % feats);
    float mean = red[2 * feats + j], istd = red[3 * feats + j];
    float out = (v[idx] - ms[j] * mean) * istd * w[j] + b[j];
    if (prelu_a) {
        float a = *prelu_a;
        out = (out >= 0.0f) ? out : a * out;
    }
    v[idx] = out;
}

// ---------------- pooling ----------------
__global__ void pool_scatter(const float* __restrict__ h, const int* __restrict__ batch,
                             float* __restrict__ pooled) {
    long idx = (long)blockIdx.x * blockDim.x + threadIdx.x;
    if (idx >= (long)NNODE * H) return;
    int i = (int)(idx >> 6), j = (int)(idx & 63);
    atomicAdd(&pooled[(long)batch[i] * H + j], h[idx]);
}

// ---------------- straight-through gumbel -> adjacency ----------------
__global__ void adj_kernel(const float* __restrict__ logits, const float* __restrict__ gum,
                           float* __restrict__ out) {
    int t = blockIdx.x * blockDim.x + threadIdx.x;
    if (t >= NB * NP) return;
    int bb = t / NP, p = t % NP;
    float a0 = logits[(long)bb * OUTD + 2 * p]     + gum[(long)bb * OUTD + 2 * p];
    float a1 = logits[(long)bb * OUTD + 2 * p + 1] + gum[(long)bb * OUTD + 2 * p + 1];
    float val = (a0 >= a1) ? 1.0f : 0.0f;   // argmax ties -> index 0
    // p -> (i,j) upper-triangular (k=1) row-major
    int i = 0, rem = p;
    while (rem >= 63 - i) { rem -= 63 - i; ++i; }
    int j = i + 1 + rem;
    out[(long)bb * 4096 + i * 64 + j] = val;
    out[(long)bb * 4096 + j * 64 + i] = val;
}

// ---------------- host launcher ----------------
static inline dim3 g1(long n, int b) { return dim3((unsigned)((n + b - 1) / b)); }

static inline void launch_gemm(const float* A, const float* W, const float* bias, float* out,
                               int M, int K, int Ncol, hipStream_t stream) {
    dim3 grid((M + 63) / 64, Ncol / 64);
    size_t shmem = (size_t)K * 64 * sizeof(float);
    wmma_gemm_bias<<<grid, 128, shmem, stream>>>(A, W, bias, out, M, K, Ncol);
}

extern "C" void kernel_launch(void* const* d_in, const int* in_sizes, int n_in,
                              void* d_out, int out_size, void* d_ws, size_t ws_size,
                              hipStream_t stream) {
    const float* x     = (const float*)d_in[0];
    const float* Wl[3] = {(const float*)d_in[1],  (const float*)d_in[8],  (const float*)d_in[15]};
    const float* bl[3] = {(const float*)d_in[2],  (const float*)d_in[9],  (const float*)d_in[16]};
    const float* Wr[3] = {(const float*)d_in[3],  (const float*)d_in[10], (const float*)d_in[17]};
    const float* br[3] = {(const float*)d_in[4],  (const float*)d_in[11], (const float*)d_in[18]};
    const float* at[3] = {(const float*)d_in[5],  (const float*)d_in[12], (const float*)d_in[19]};
    const float* bo[3] = {(const float*)d_in[6],  (const float*)d_in[13], (const float*)d_in[20]};
    const float* pr[3] = {(const float*)d_in[7],  (const float*)d_in[14], (const float*)d_in[21]};
    const float* gn_w  = (const float*)d_in[22];
    const float* gn_b  = (const float*)d_in[23];
    const float* gn_ms = (const float*)d_in[24];
    const float* fc_W  = (const float*)d_in[25];
    const float* fc_b  = (const float*)d_in[26];
    const float* mu_W  = (const float*)d_in[27];
    const float* mu_b  = (const float*)d_in[28];
    const float* d0_W  = (const float*)d_in[31];
    const float* d0_b  = (const float*)d_in[32];
    const float* d1_W  = (const float*)d_in[33];
    const float* d1_b  = (const float*)d_in[34];
    const float* d2_W  = (const float*)d_in[35];
    const float* d2_b  = (const float*)d_in[36];
    const float* dgn_w = (const float*)d_in[37];
    const float* dgn_b = (const float*)d_in[38];
    const float* dgn_ms= (const float*)d_in[39];
    const float* dprelu= (const float*)d_in[40];
    const float* gum   = (const float*)d_in[41];
    const int*   ei    = (const int*)d_in[42];
    const int*   batch = (const int*)d_in[43];
    float* out = (float*)d_out;

    // workspace carve-up (floats)
    float* f = (float*)d_ws;
    long o = 0;
    float* h      = f + o; o += (long)NNODE * H;
    float* hl     = f + o; o += (long)NNODE * H;
    float* hr     = f + o; o += (long)NNODE * H;
    float* acc    = f + o; o += (long)NNODE * H;
    float* denom  = f + o; o += NNODE;
    float* score  = f + o; o += EE;
    float* ex     = f + o; o += EE;
    float* red    = f + o; o += 1024;                // 4*256
    float* pooled = f + o; o += (long)NB * H;
    float* bufB   = f + o; o += (long)NB * H;
    float* db0    = f + o; o += (long)NB * HD;
    float* db1    = f + o; o += (long)NB * HD;
    float* logits = f + o; o += (long)NB * OUTD;
    int*   smax   = (int*)(f + o);                   // NNODE ints

    const int T = 256;
    const long NH = (long)NNODE * H;

    const float* hin = x; int Kin = 128;
    for (int l = 0; l < 3; ++l) {
        launch_gemm(hin, Wl[l], bl[l], hl, NNODE, Kin, H, stream);
        launch_gemm(hin, Wr[l], br[l], hr, NNODE, Kin, H, stream);

        zero_f<<<g1(NNODE, T), T, 0, stream>>>(denom, NNODE);
        zero_f<<<g1(NH, T),    T, 0, stream>>>(acc, NH);
        zero_f<<<g1(1024, T),  T, 0, stream>>>(red, 1024);
        init_i<<<g1(NNODE, T), T, 0, stream>>>(smax, NNODE, (int)0x80000000);

        edge_score  <<<g1(EE, T), T, 0, stream>>>(hl, hr, at[l], ei, score, smax);
        edge_exp    <<<g1(EE, T), T, 0, stream>>>(score, smax, ei, ex, denom);
        edge_scatter<<<g1(EE, T), T, 0, stream>>>(ex, hl, ei, acc);

        node_combine<<<g1(NH, T), T, 0, stream>>>(acc, denom, bo[l], h);
        reduce_sums <<<256, H, 0, stream>>>(h, NNODE, H, red);
        norm_stats  <<<1,   H, 0, stream>>>(red, NNODE, H, gn_ms);
        normalize_k <<<g1(NH, T), T, 0, stream>>>(h, NNODE, H, gn_w, gn_b, gn_ms, red, pr[l]);
        hin = h; Kin = H;
    }

    // ---- pooling + norm ----
    zero_f<<<g1((long)NB * H, T), T, 0, stream>>>(pooled, (long)NB * H);
    pool_scatter<<<g1(NH, T), T, 0, stream>>>(h, batch, pooled);
    zero_f<<<g1(1024, T), T, 0, stream>>>(red, 1024);
    reduce_sums<<<128, H, 0, stream>>>(pooled, NB, H, red);
    norm_stats <<<1,   H, 0, stream>>>(red, NB, H, gn_ms);
    normalize_k<<<g1((long)NB * H, T), T, 0, stream>>>(pooled, NB, H, gn_w, gn_b, gn_ms, red, nullptr);

    // ---- encoder head: xg = pooled@fc, z = xg@mu ----
    launch_gemm(pooled, fc_W, fc_b, bufB, NB, H, H, stream);
    float* z = hl;  // reuse scratch
    launch_gemm(bufB, mu_W, mu_b, z, NB, H, H, stream);

    // ---- decoder ----
    launch_gemm(z, d0_W, d0_b, db0, NB, H, HD, stream);
    zero_f<<<g1(1024, T), T, 0, stream>>>(red, 1024);
    reduce_sums<<<128, HD, 0, stream>>>(db0, NB, HD, red);
    norm_stats <<<1,   HD, 0, stream>>>(red, NB, HD, dgn_ms);
    normalize_k<<<g1((long)NB * HD, T), T, 0, stream>>>(db0, NB, HD, dgn_w, dgn_b, dgn_ms, red, dprelu);

    launch_gemm(db0, d1_W, d1_b, db1, NB, HD, HD, stream);
    zero_f<<<g1(1024, T), T, 0, stream>>>(red, 1024);
    reduce_sums<<<128, HD, 0, stream>>>(db1, NB, HD, red);
    norm_stats <<<1,   HD, 0, stream>>>(red, NB, HD, dgn_ms);
    normalize_k<<<g1((long)NB * HD, T), T, 0, stream>>>(db1, NB, HD, dgn_w, dgn_b, dgn_ms, red, dprelu);

    launch_gemm(db1, d2_W, d2_b, logits, NB, HD, OUTD, stream);

    // ---- adjacency output ----
    zero_f<<<g1((long)NB * 4096, T), T, 0, stream>>>(out, (long)NB * 4096);
    adj_kernel<<<g1((long)NB * NP, T), T, 0, stream>>>(logits, gum, out);
}